// ComplexFaberConv_64647847739450
// MI455X (gfx1250) — compile-verified
//
#include <hip/hip_runtime.h>

#define NN 100000
#define NE 640000
#define DD 128
#define LROW 132   // LDS row pitch (floats): 16B-aligned rows, conflict-free frag reads

typedef float v2f __attribute__((ext_vector_type(2)));
typedef float v8f __attribute__((ext_vector_type(8)));

// ---------------------------------------------------------------- zero scratch
__global__ void __launch_bounds__(256) zero_kernel(float4* __restrict__ p, int n4) {
    int i = blockIdx.x * blockDim.x + threadIdx.x;
    if (i < n4) p[i] = make_float4(0.f, 0.f, 0.f, 0.f);
}

// ------------------------------------------- fold K=3 hops into effective W/b
// WrE[o*128+d] = sum_k s_k * W_real[k,o,d]   (s = 1, .5, .25)
// bR[o] = s@(b_real - b_imag), bI[o] = s@(b_real + b_imag)
__global__ void __launch_bounds__(256) prep_weights(
    const float* __restrict__ Wr, const float* __restrict__ Wi,
    const float* __restrict__ br, const float* __restrict__ bi,
    float* __restrict__ WrE, float* __restrict__ WiE,
    float* __restrict__ bR, float* __restrict__ bI) {
    int i = blockIdx.x * blockDim.x + threadIdx.x;
    if (i >= DD * DD) return;
    WrE[i] = Wr[i] + 0.5f * Wr[DD*DD + i] + 0.25f * Wr[2*DD*DD + i];
    WiE[i] = Wi[i] + 0.5f * Wi[DD*DD + i] + 0.25f * Wi[2*DD*DD + i];
    if (i < DD) {
        float brs = br[i] + 0.5f * br[DD + i] + 0.25f * br[2*DD + i];
        float bis = bi[i] + 0.5f * bi[DD + i] + 0.25f * bi[2*DD + i];
        bR[i] = brs - bis;
        bI[i] = brs + bis;
    }
}

// ---------------------------------------------------------------- degrees
__global__ void __launch_bounds__(256) degree_kernel(
    const int* __restrict__ row, const int* __restrict__ col,
    float* __restrict__ degO, float* __restrict__ degI) {
    int e = blockIdx.x * blockDim.x + threadIdx.x;
    if (e < NE) {
        atomicAdd(degO + row[e], 1.0f);
        atomicAdd(degI + col[e], 1.0f);
    }
}

// deg -> deg^-0.25 in place (0 stays 0)
__global__ void __launch_bounds__(256) degnorm_kernel(float* __restrict__ deg, int n) {
    int i = blockIdx.x * blockDim.x + threadIdx.x;
    if (i < n) {
        float v = deg[i];
        deg[i] = (v > 0.f) ? powf(v, -0.25f) : 0.f;
    }
}

// ------------------------------------------------- SPMM scatter (1 wave/edge)
// yr [row] += val*x_real[col]
// yrt[col] += val*x_real[row]
// ysi[row] += val*x_imag[col]   (y_i)
// ysi[col] += val*x_imag[row]   (y_it, shares accumulator)
__global__ void __launch_bounds__(256) scatter_kernel(
    const int* __restrict__ row, const int* __restrict__ col,
    const float* __restrict__ xr, const float* __restrict__ xi,
    const float* __restrict__ dn,   // [0,NN)=dn_out, [NN,2NN)=dn_in
    float* __restrict__ yr, float* __restrict__ yrt, float* __restrict__ ysi) {
    int tid  = blockIdx.x * blockDim.x + threadIdx.x;
    int e    = tid >> 5;
    if (e >= NE) return;
    int lane = tid & 31;
    int r = row[e], c = col[e];
    float val = dn[r] * dn[NN + c];
    int f = lane << 2;                      // 32 lanes * 4 floats = 128 features
    const float4 xrc = *(const float4*)(xr + c * DD + f);
    const float4 xrr = *(const float4*)(xr + r * DD + f);
    const float4 xic = *(const float4*)(xi + c * DD + f);
    const float4 xir = *(const float4*)(xi + r * DD + f);
    float* p;
    p = yr + r * DD + f;
    atomicAdd(p+0, val*xrc.x); atomicAdd(p+1, val*xrc.y);
    atomicAdd(p+2, val*xrc.z); atomicAdd(p+3, val*xrc.w);
    p = yrt + c * DD + f;
    atomicAdd(p+0, val*xrr.x); atomicAdd(p+1, val*xrr.y);
    atomicAdd(p+2, val*xrr.z); atomicAdd(p+3, val*xrr.w);
    p = ysi + r * DD + f;
    atomicAdd(p+0, val*xic.x); atomicAdd(p+1, val*xic.y);
    atomicAdd(p+2, val*xic.z); atomicAdd(p+3, val*xic.w);
    p = ysi + c * DD + f;
    atomicAdd(p+0, val*xir.x); atomicAdd(p+1, val*xir.y);
    atomicAdd(p+2, val*xir.z); atomicAdd(p+3, val*xir.w);
}

// ----------------------------------------------------- WMMA fp32 GEMM epilogue
// Per wave: one 16x16 tile of total_real and total_imag.
//   total_real = A@WrE^T - B@WiE^T + bR,   A = 0.5(yr+yrt), B = 0.5*ysi
//   total_imag = C@WiE^T + B@WrE^T + bI,   C = yr
// A/B/C tiles are staged once per block into LDS (pre-fused), so the inner
// loop is 3x ds_load_b64 + 2x global_load_b64 (L2-hot weights) + 4x WMMA.
// Fragment layout (ISA 7.12.2, 32-bit, wave32):
//   A frag (16x4): lane<16 -> row=lane, K={k0,k0+1}; lane>=16 -> row=lane-16, K={k0+2,k0+3}
//   B frag (4x16): lane<16 -> col=lane, K={k0,k0+1}; lane>=16 -> col=lane-16, K={k0+2,k0+3}
//   C/D (16x16): VGPR v, lane -> row = v + 8*(lane>>4), col = lane&15
__global__ void __launch_bounds__(256) gemm_kernel(
    const float* __restrict__ yr, const float* __restrict__ yrt,
    const float* __restrict__ ysi,
    const float* __restrict__ WrE, const float* __restrict__ WiE,
    const float* __restrict__ bR, const float* __restrict__ bI,
    float* __restrict__ outR, float* __restrict__ outI) {
    __shared__ float sA[16 * LROW];
    __shared__ float sB[16 * LROW];
    __shared__ float sC[16 * LROW];

    const int t = threadIdx.x;

    // ---- stage node tile (16 rows x 128) once per block, pre-fusing A/B/C
    {
        const int r  = t >> 4;            // 0..15
        const int cg = (t & 15) << 3;     // 0,8,...,120
        const int gRow = blockIdx.x * 16 + r;
        const float* yrG  = yr  + gRow * DD + cg;
        const float* yrtG = yrt + gRow * DD + cg;
        const float* ysiG = ysi + gRow * DD + cg;
        float* aL = sA + r * LROW + cg;
        float* bL = sB + r * LROW + cg;
        float* cL = sC + r * LROW + cg;
#pragma unroll
        for (int j = 0; j < 2; ++j) {
            float4 vr = *(const float4*)(yrG  + 4 * j);
            float4 vt = *(const float4*)(yrtG + 4 * j);
            float4 vi = *(const float4*)(ysiG + 4 * j);
            float4 va = make_float4(0.5f * (vr.x + vt.x), 0.5f * (vr.y + vt.y),
                                    0.5f * (vr.z + vt.z), 0.5f * (vr.w + vt.w));
            float4 vb = make_float4(0.5f * vi.x, 0.5f * vi.y,
                                    0.5f * vi.z, 0.5f * vi.w);
            *(float4*)(aL + 4 * j) = va;
            *(float4*)(bL + 4 * j) = vb;
            *(float4*)(cL + 4 * j) = vr;
        }
    }
    __syncthreads();

    const int wave = t >> 5;              // 8 waves -> 8 col tiles of 16
    const int lane = t & 31;
    const int half = lane >> 4;           // K sub-pair selector
    const int l15  = lane & 15;
    const int o = wave * 16 + l15;        // output col this lane feeds into B frag

    const float* aP = sA + l15 * LROW;    // A frag row for this lane
    const float* bP = sB + l15 * LROW;
    const float* cP = sC + l15 * LROW;
    const float* wrP = WrE + o * DD;      // W^T[d][o] = WrE[o*DD + d]
    const float* wiP = WiE + o * DD;

    v8f accAR = {};   // A @ Wr^T
    v8f accBI = {};   // B @ Wi^T
    v8f accCI = {};   // C @ Wi^T
    v8f accBR = {};   // B @ Wr^T

#pragma unroll 8
    for (int k0 = 0; k0 < DD; k0 += 4) {
        const int kk = k0 + half * 2;
        v2f aA  = *(const v2f*)(aP + kk);
        v2f aB  = *(const v2f*)(bP + kk);
        v2f aC  = *(const v2f*)(cP + kk);
        v2f bWr = *(const v2f*)(wrP + kk);
        v2f bWi = *(const v2f*)(wiP + kk);
        accAR = __builtin_amdgcn_wmma_f32_16x16x4_f32(false, aA, false, bWr,
                                                      (short)0, accAR, false, false);
        accBI = __builtin_amdgcn_wmma_f32_16x16x4_f32(false, aB, false, bWi,
                                                      (short)0, accBI, false, false);
        accCI = __builtin_amdgcn_wmma_f32_16x16x4_f32(false, aC, false, bWi,
                                                      (short)0, accCI, false, false);
        accBR = __builtin_amdgcn_wmma_f32_16x16x4_f32(false, aB, false, bWr,
                                                      (short)0, accBR, false, false);
    }

    // bias depends only on the output column = o for this lane
    const float biasR = bR[o];
    const float biasI = bI[o];
    v8f accR = accAR - accBI + biasR;
    v8f accI = accCI + accBR + biasI;

#pragma unroll
    for (int r = 0; r < 8; ++r) {
        int rowOut = blockIdx.x * 16 + r + 8 * half;
        outR[rowOut * DD + o] = accR[r];
        outI[rowOut * DD + o] = accI[r];
    }
}

// ---------------------------------------------------------------------- launch
extern "C" void kernel_launch(void* const* d_in, const int* in_sizes, int n_in,
                              void* d_out, int out_size, void* d_ws, size_t ws_size,
                              hipStream_t stream) {
    const float* x_real = (const float*)d_in[0];
    const float* x_imag = (const float*)d_in[1];
    const float* W_real = (const float*)d_in[2];
    const float* W_imag = (const float*)d_in[3];
    const float* b_real = (const float*)d_in[4];
    const float* b_imag = (const float*)d_in[5];
    const int*   eidx   = (const int*)d_in[6];
    const int* row = eidx;
    const int* col = eidx + NE;

    // workspace layout (floats):
    float* ws  = (float*)d_ws;
    float* deg = ws;                  // 2*NN  (deg_out | deg_in -> dn_out | dn_in)
    float* yr  = ws + 2 * NN;         // NN*DD
    float* yrt = yr + NN * DD;        // NN*DD
    float* ysi = yrt + NN * DD;       // NN*DD
    float* WrE = ysi + NN * DD;       // DD*DD
    float* WiE = WrE + DD * DD;       // DD*DD
    float* bR  = WiE + DD * DD;       // DD
    float* bI  = bR + DD;             // DD

    float* outR = (float*)d_out;
    float* outI = outR + NN * DD;

    // 1) zero dynamic accumulators (every call: graph replays reuse ws)
    const int n4 = (2 * NN + 3 * NN * DD) / 4;
    zero_kernel<<<(n4 + 255) / 256, 256, 0, stream>>>((float4*)ws, n4);

    // 2) effective weights/biases
    prep_weights<<<(DD * DD + 255) / 256, 256, 0, stream>>>(
        W_real, W_imag, b_real, b_imag, WrE, WiE, bR, bI);

    // 3) degrees
    degree_kernel<<<(NE + 255) / 256, 256, 0, stream>>>(row, col, deg, deg + NN);

    // 4) deg^-0.25 in place over both halves
    degnorm_kernel<<<(2 * NN + 255) / 256, 256, 0, stream>>>(deg, 2 * NN);

    // 5) scatter: one wave per edge -> 640000 waves
    scatter_kernel<<<(NE * 32) / 256, 256, 0, stream>>>(
        row, col, x_real, x_imag, deg, yr, yrt, ysi);

    // 6) WMMA GEMM: N/16 = 6250 row tiles, 8 waves cover the 128 out cols
    gemm_kernel<<<NN / 16, 256, 0, stream>>>(
        yr, yrt, ysi, WrE, WiE, bR, bI, outR, outI);
}